// RotateChordVarLen_25357486916298
// MI455X (gfx1250) — compile-verified
//
#include <hip/hip_runtime.h>

// RotateChordVarLen: ragged per-segment circular roll, track i rolled by
// 2^(i-1) within each segment. Pure gather: 0 FLOPs, ~268 MB traffic
// -> ~11.5 us HBM floor at 23.3 TB/s.
// Strategy (MI455X): 1 float4 per lane; contiguous 512B stores per wave32;
// NT stores keep the 134MB input resident in the 192MB L2 so the strided
// 64B gather reads hit L2 instead of amplifying HBM traffic.

#define NSEG 8          // B = 8 segments (reference LENGTHS)
#define ROWF4 64        // 256 floats per row = 64 float4

typedef float v4f __attribute__((ext_vector_type(4)));  // native vector: OK for NT builtins

__global__ __launch_bounds__(256)
void RotateChordVarLen_kernel(const float* __restrict__ x,
                              const int* __restrict__ lengths,
                              float* __restrict__ out,
                              int T) {
    int tid = blockIdx.x * blockDim.x + threadIdx.x;
    int total = T * ROWF4;
    if (tid >= total) return;

    int row = tid >> 6;     // output token row
    int q   = tid & 63;     // float4 index within the row (track = q>>2)
    int track = q >> 2;

    // Segment search: unrolled prefix-sum scan over 8 lengths.
    // lengths[] reads are uniform -> scalar loads.
    int start = 0;
    int len   = lengths[0];
    int acc   = len;
#pragma unroll
    for (int i = 1; i < NSEG; ++i) {
        int li = lengths[i];
        if (row >= acc) { start = acc; len = li; }
        acc += li;
    }

    unsigned pos   = (unsigned)(row - start);
    unsigned shift = (track == 0) ? 0u : (1u << (track - 1));
    unsigned len_u = (unsigned)len;

    unsigned s = pos + shift;
    if (s >= len_u) {               // common wrap: one subtract
        s -= len_u;
        if (s >= len_u) s %= len_u; // rare: shift >= len (small segments)
    }

    // src/dst indices in float4 units; column offset is exactly q.
    size_t src_idx = (size_t)(start + (int)s) * ROWF4 + (size_t)q;
    size_t dst_idx = (size_t)tid;

    v4f v = ((const v4f*)x)[src_idx];                       // RT load: cache in L2
    __builtin_nontemporal_store(v, ((v4f*)out) + dst_idx);  // NT 128-bit store
}

extern "C" void kernel_launch(void* const* d_in, const int* in_sizes, int n_in,
                              void* d_out, int out_size, void* d_ws, size_t ws_size,
                              hipStream_t stream) {
    const float* x       = (const float*)d_in[0];
    const int*   lengths = (const int*)d_in[1];
    float*       out     = (float*)d_out;

    int T = in_sizes[0] / 256;       // D_MODEL = 256
    int total = T * ROWF4;           // float4 work items
    int threads = 256;               // 8 waves per workgroup (wave32)
    int blocks = (total + threads - 1) / threads;

    RotateChordVarLen_kernel<<<blocks, threads, 0, stream>>>(x, lengths, out, T);
}